// DensityEstimator_67791763800794
// MI455X (gfx1250) — compile-verified
//
#include <hip/hip_runtime.h>

// DensityEstimator: per-channel 1->3->3->3->1 monotone MLP, evaluated at x+0.5
// and x-0.5, differenced. Bandwidth/transcendental bound (~101 MB @ 23.3 TB/s
// => ~4.3us floor). Per-channel 3x3 matmuls are too small for WMMA (>=86% MAC
// waste + a cross-lane transpose per layer due to the elementwise tanh gate);
// dense VALU FMA + gfx1250 hardware transcendentals (v_tanh_f32/v_exp_f32/
// v_rcp_f32) is the optimal compute path. CDNA5 data movement: the 33KB
// pre-transformed parameter block is staged into LDS per block with async
// global->LDS DMA (ASYNCcnt), overlapped with the x-tile global loads. The
// DMA uses the instruction's 24-bit INST_OFFSET (applied to both LDS and
// global addresses) so the 43 copies share 3 base VGPRs with zero address ALU.

#define CH    192
#define ITEMS 8

// Workspace slot layout (SoA): P[slot*CH + c]. Total 43 slots * 192 ch * 4B = 33,024 B.
#define S_W0 0    // softplus(H0): 3
#define S_W1 3    // softplus(H1): 9  (flat i*3+p)
#define S_W2 12   // softplus(H2): 9
#define S_W3 21   // softplus(H3): 3
#define S_B0 24   // b0: 3
#define S_B1 27   // b1: 3
#define S_B2 30   // b2: 3
#define S_B3 33   // b3: 1
#define S_T0 34   // tanh(a0): 3
#define S_T1 37   // tanh(a1): 3
#define S_T2 40   // tanh(a2): 3
#define NSLOT 43

#define LOG2E  1.4426950408889634f
#define LN2    0.6931471805599453f

__device__ __forceinline__ float exp2_f(float v) { return __builtin_amdgcn_exp2f(v); }
__device__ __forceinline__ float rcp_f(float v)  { return __builtin_amdgcn_rcpf(v); }
__device__ __forceinline__ float log2_f(float v) { return __builtin_amdgcn_logf(v); }

__device__ __forceinline__ float tanh_f(float v) {
#if __has_builtin(__builtin_amdgcn_tanhf)
  return __builtin_amdgcn_tanhf(v);           // gfx1250 v_tanh_f32 (1 trans op)
#else
  // tanh(x) = 1 - 2/(1 + e^(2x)); saturates correctly as exp2 -> 0 / inf
  float e = exp2_f(v * (2.0f * LOG2E));
  return 1.0f - 2.0f * rcp_f(1.0f + e);
#endif
}

__device__ __forceinline__ float sigmoid_f(float v) {
  return rcp_f(1.0f + exp2_f(-v * LOG2E));    // exact at +/- inf
}

__device__ __forceinline__ float softplus_f(float v) {
  // softplus(x) = max(x,0) + ln(1 + e^-|x|), overflow-free
  float lp = LN2 * log2_f(1.0f + exp2_f(-fabsf(v) * LOG2E));
  return fmaxf(v, 0.0f) + lp;
}

// ---------------------------------------------------------------------------
// Kernel 1: hoist softplus(H*) / tanh(a*) out of the 12.6M-element loop.
// One thread per channel; SoA layout so de_main's staging DMA coalesces.
// ---------------------------------------------------------------------------
__global__ __launch_bounds__(CH) void de_prep(
    const float* __restrict__ a0, const float* __restrict__ a1, const float* __restrict__ a2,
    const float* __restrict__ b0, const float* __restrict__ b1, const float* __restrict__ b2,
    const float* __restrict__ b3,
    const float* __restrict__ H0, const float* __restrict__ H1,
    const float* __restrict__ H2, const float* __restrict__ H3,
    float* __restrict__ P, int channels)
{
  int c = blockIdx.x * blockDim.x + threadIdx.x;
  if (c >= channels) return;
#pragma unroll
  for (int j = 0; j < 3; ++j) P[(S_W0 + j) * CH + c] = softplus_f(H0[c * 3 + j]);
#pragma unroll
  for (int j = 0; j < 9; ++j) P[(S_W1 + j) * CH + c] = softplus_f(H1[c * 9 + j]);
#pragma unroll
  for (int j = 0; j < 9; ++j) P[(S_W2 + j) * CH + c] = softplus_f(H2[c * 9 + j]);
#pragma unroll
  for (int j = 0; j < 3; ++j) P[(S_W3 + j) * CH + c] = softplus_f(H3[c * 3 + j]);
#pragma unroll
  for (int j = 0; j < 3; ++j) P[(S_B0 + j) * CH + c] = b0[c * 3 + j];
#pragma unroll
  for (int j = 0; j < 3; ++j) P[(S_B1 + j) * CH + c] = b1[c * 3 + j];
#pragma unroll
  for (int j = 0; j < 3; ++j) P[(S_B2 + j) * CH + c] = b2[c * 3 + j];
  P[S_B3 * CH + c] = b3[c];
#pragma unroll
  for (int j = 0; j < 3; ++j) P[(S_T0 + j) * CH + c] = tanh_f(a0[c * 3 + j]);
#pragma unroll
  for (int j = 0; j < 3; ++j) P[(S_T1 + j) * CH + c] = tanh_f(a1[c * 3 + j]);
#pragma unroll
  for (int j = 0; j < 3; ++j) P[(S_T2 + j) * CH + c] = tanh_f(a2[c * 3 + j]);
}

// ---------------------------------------------------------------------------
// Per-channel parameters held in registers.
// ---------------------------------------------------------------------------
struct DP {
  float W0[3], W1[9], W2[9], W3[3];
  float B0[3], B1[3], B2[3], B3;
  float T0[3], T1[3], T2[3];
};

// Both CDF passes (x+0.5, x-0.5) interleaved for ILP across transcendentals.
__device__ __forceinline__ float eval_pdf(float xin, const DP& P) {
  float u[2] = { xin + 0.5f, xin - 0.5f };
  float y[2][3], z[2][3], r[2];
  // layer 0: 1 -> 3, gate
#pragma unroll
  for (int s = 0; s < 2; ++s)
#pragma unroll
    for (int p = 0; p < 3; ++p) {
      float t = fmaf(u[s], P.W0[p], P.B0[p]);
      y[s][p] = fmaf(P.T0[p], tanh_f(t), t);
    }
  // layer 1: 3 -> 3, gate
#pragma unroll
  for (int s = 0; s < 2; ++s)
#pragma unroll
    for (int p = 0; p < 3; ++p) {
      float t = P.B1[p];
#pragma unroll
      for (int i = 0; i < 3; ++i) t = fmaf(y[s][i], P.W1[i * 3 + p], t);
      z[s][p] = fmaf(P.T1[p], tanh_f(t), t);
    }
  // layer 2: 3 -> 3, gate
#pragma unroll
  for (int s = 0; s < 2; ++s)
#pragma unroll
    for (int p = 0; p < 3; ++p) {
      float t = P.B2[p];
#pragma unroll
      for (int i = 0; i < 3; ++i) t = fmaf(z[s][i], P.W2[i * 3 + p], t);
      y[s][p] = fmaf(P.T2[p], tanh_f(t), t);
    }
  // layer 3: 3 -> 1, sigmoid
#pragma unroll
  for (int s = 0; s < 2; ++s) {
    float t = P.B3;
#pragma unroll
    for (int i = 0; i < 3; ++i) t = fmaf(y[s][i], P.W3[i], t);
    r[s] = sigmoid_f(t);
  }
  return r[0] - r[1];
}

// One async b32 copy per slot: same 24-bit immediate applies to both the LDS
// dest (VDST + offset) and global src (VADDR + offset) -> zero address ALU.
#define ACP(OFF) asm volatile("global_load_async_to_lds_b32 %0, %1, off offset:" #OFF \
                              :: "v"(l), "v"(g) : "memory")

// ---------------------------------------------------------------------------
// Kernel 2: thread = channel, ITEMS batch rows per thread. The 33KB parameter
// block is DMA'd global->LDS (async, ASYNCcnt) while the x-tile global loads
// are issued, then read to registers via ds_load(_2addr)_b32 with immediate
// offsets. x/out accesses are 768B-contiguous per block row.
// ---------------------------------------------------------------------------
__global__ __launch_bounds__(CH) void de_main(
    const float* __restrict__ x, const float* __restrict__ Pws,
    float* __restrict__ out, int batch)
{
  const int c = threadIdx.x;            // channel 0..191
  const int row0 = blockIdx.x * ITEMS;

  __shared__ float sP[NSLOT * CH];      // 33,024 B of 320KB WGP LDS

  // gfx1250 async global->LDS DMA: each wave copies its own 32 channels x 43
  // slots (128B per instruction, per-lane addresses). Offsets 0..42*768.
  {
    unsigned long long g = (unsigned long long)(Pws + c);
    unsigned l = (unsigned)(size_t)(&sP[c]);   // low 32 bits = LDS byte offset
    ACP(0);     ACP(768);   ACP(1536);  ACP(2304);  ACP(3072);  ACP(3840);
    ACP(4608);  ACP(5376);  ACP(6144);  ACP(6912);  ACP(7680);  ACP(8448);
    ACP(9216);  ACP(9984);  ACP(10752); ACP(11520); ACP(12288); ACP(13056);
    ACP(13824); ACP(14592); ACP(15360); ACP(16128); ACP(16896); ACP(17664);
    ACP(18432); ACP(19200); ACP(19968); ACP(20736); ACP(21504); ACP(22272);
    ACP(23040); ACP(23808); ACP(24576); ACP(25344); ACP(26112); ACP(26880);
    ACP(27648); ACP(28416); ACP(29184); ACP(29952); ACP(30720); ACP(31488);
    ACP(32256);
  }

  // Overlap: issue the x-tile loads while the async param DMA is in flight.
  float xv[ITEMS];
  const float* xp = x + (size_t)row0 * CH + c;
  const bool full = (row0 + ITEMS) <= batch;
  if (full) {
#pragma unroll
    for (int i = 0; i < ITEMS; ++i) xv[i] = xp[i * CH];   // imm offsets i*768
  } else {
#pragma unroll
    for (int i = 0; i < ITEMS; ++i) xv[i] = (row0 + i < batch) ? xp[i * CH] : 0.0f;
  }

  asm volatile("s_wait_asynccnt 0x0" ::: "memory");
  __syncthreads();

  DP P;                                 // ds loads, all-immediate offsets
#pragma unroll
  for (int j = 0; j < 3; ++j) P.W0[j] = sP[(S_W0 + j) * CH + c];
#pragma unroll
  for (int j = 0; j < 9; ++j) P.W1[j] = sP[(S_W1 + j) * CH + c];
#pragma unroll
  for (int j = 0; j < 9; ++j) P.W2[j] = sP[(S_W2 + j) * CH + c];
#pragma unroll
  for (int j = 0; j < 3; ++j) P.W3[j] = sP[(S_W3 + j) * CH + c];
#pragma unroll
  for (int j = 0; j < 3; ++j) P.B0[j] = sP[(S_B0 + j) * CH + c];
#pragma unroll
  for (int j = 0; j < 3; ++j) P.B1[j] = sP[(S_B1 + j) * CH + c];
#pragma unroll
  for (int j = 0; j < 3; ++j) P.B2[j] = sP[(S_B2 + j) * CH + c];
  P.B3 = sP[S_B3 * CH + c];
#pragma unroll
  for (int j = 0; j < 3; ++j) P.T0[j] = sP[(S_T0 + j) * CH + c];
#pragma unroll
  for (int j = 0; j < 3; ++j) P.T1[j] = sP[(S_T1 + j) * CH + c];
#pragma unroll
  for (int j = 0; j < 3; ++j) P.T2[j] = sP[(S_T2 + j) * CH + c];

  // Single copy of the compute body (no branch duplication of eval_pdf).
  float pv[ITEMS];
#pragma unroll
  for (int i = 0; i < ITEMS; ++i) pv[i] = eval_pdf(xv[i], P);

  float* op = out + (size_t)row0 * CH + c;
  if (full) {
#pragma unroll
    for (int i = 0; i < ITEMS; ++i) op[i * CH] = pv[i];
  } else {
#pragma unroll
    for (int i = 0; i < ITEMS; ++i)
      if (row0 + i < batch) op[i * CH] = pv[i];
  }
}

// ---------------------------------------------------------------------------
extern "C" void kernel_launch(void* const* d_in, const int* in_sizes, int n_in,
                              void* d_out, int out_size, void* d_ws, size_t ws_size,
                              hipStream_t stream)
{
  const float* x  = (const float*)d_in[0];
  const float* a0 = (const float*)d_in[1];
  const float* a1 = (const float*)d_in[2];
  const float* a2 = (const float*)d_in[3];
  const float* b0 = (const float*)d_in[4];
  const float* b1 = (const float*)d_in[5];
  const float* b2 = (const float*)d_in[6];
  const float* b3 = (const float*)d_in[7];
  const float* H0 = (const float*)d_in[8];
  const float* H1 = (const float*)d_in[9];
  const float* H2 = (const float*)d_in[10];
  const float* H3 = (const float*)d_in[11];
  float* P   = (float*)d_ws;            // needs NSLOT*CH*4 = 33,024 bytes
  float* out = (float*)d_out;

  const int batch = in_sizes[0] / CH;   // 65536

  de_prep<<<1, CH, 0, stream>>>(a0, a1, a2, b0, b1, b2, b3,
                                H0, H1, H2, H3, P, CH);
  const int nblk = (batch + ITEMS - 1) / ITEMS;
  de_main<<<nblk, CH, 0, stream>>>(x, P, out, batch);
}